// Demosaic_15049565405687
// MI455X (gfx1250) — compile-verified
//
#include <hip/hip_runtime.h>

// Malvar demosaic, bandwidth-bound (~537 MB @ 23.3 TB/s => ~23us floor).
// Strategy: async global->LDS tile staging (CDNA5 ASYNCcnt path), parity-
// specialized 2x2-quad compute from LDS, float2 coalesced stores.

#define H_DIM 2048
#define W_DIM 2048
#define TILE 32
#define PADR 2
#define LSX  36               // LDS row stride = TILE + 2*PADR (even -> 8B aligned float2)
#define LSY  36
#define NTHREADS 256

#if __has_builtin(__builtin_amdgcn_global_load_async_to_lds_b32) && \
    __has_builtin(__builtin_amdgcn_s_wait_asynccnt)
#define USE_ASYNC_LDS 1
#endif

#define AS_GLOBAL __attribute__((address_space(1)))
#define AS_SHARED __attribute__((address_space(3)))

__device__ __forceinline__ int reflect2(int i, int n) {
  // jnp.pad 'reflect' with pad<=2: -1->1, -2->2, n->n-2, n+1->n-3
  i = (i < 0) ? -i : i;
  return (i >= n) ? (2 * n - 2 - i) : i;
}

__device__ __forceinline__ float clamp01(float v) {
  return fminf(fmaxf(v, 0.0f), 1.0f);
}

// ---- Malvar stencils, centered at w[cy][cx]; 1/8 folded into literals ----

// green at R/B site
__device__ __forceinline__ float conv_g(const float (&w)[6][6], int cy, int cx) {
  return (4.0f * w[cy][cx]
        + 2.0f * (w[cy-1][cx] + w[cy+1][cx] + w[cy][cx-1] + w[cy][cx+1])
        -        (w[cy-2][cx] + w[cy+2][cx] + w[cy][cx-2] + w[cy][cx+2])) * 0.125f;
}
// R at Gr / B at Gb (same-row neighbors of like color)
__device__ __forceinline__ float conv_col(const float (&w)[6][6], int cy, int cx) {
  return (5.0f * w[cy][cx]
        + 4.0f * (w[cy][cx-1] + w[cy][cx+1])
        -        (w[cy][cx-2] + w[cy][cx+2])
        -        (w[cy-1][cx-1] + w[cy-1][cx+1] + w[cy+1][cx-1] + w[cy+1][cx+1])
        + 0.5f * (w[cy-2][cx] + w[cy+2][cx])) * 0.125f;
}
// R at Gb / B at Gr (same-column neighbors of like color)
__device__ __forceinline__ float conv_row(const float (&w)[6][6], int cy, int cx) {
  return (5.0f * w[cy][cx]
        + 4.0f * (w[cy-1][cx] + w[cy+1][cx])
        -        (w[cy-2][cx] + w[cy+2][cx])
        -        (w[cy-1][cx-1] + w[cy-1][cx+1] + w[cy+1][cx-1] + w[cy+1][cx+1])
        + 0.5f * (w[cy][cx-2] + w[cy][cx+2])) * 0.125f;
}
// R at B site / B at R site
__device__ __forceinline__ float conv_br(const float (&w)[6][6], int cy, int cx) {
  return (6.0f * w[cy][cx]
        + 2.0f * (w[cy-1][cx-1] + w[cy-1][cx+1] + w[cy+1][cx-1] + w[cy+1][cx+1])
        - 1.5f * (w[cy-2][cx] + w[cy+2][cx] + w[cy][cx-2] + w[cy][cx+2])) * 0.125f;
}

__global__ __launch_bounds__(NTHREADS)
void demosaic_malvar_kernel(const float* __restrict__ in, float* __restrict__ out) {
  __shared__ float tile[LSY * LSX];

  const int tid = threadIdx.x;
  const int gx0 = blockIdx.x * TILE;
  const int gy0 = blockIdx.y * TILE;
  const int img = blockIdx.z;
  const size_t plane = (size_t)H_DIM * W_DIM;
  const float* __restrict__ src = in + (size_t)img * plane;

  // ---- Stage 36x36 input tile (with reflect halo) into LDS via async copies ----
  for (int i = tid; i < LSY * LSX; i += NTHREADS) {
    const int ly = i / LSX;
    const int lx = i - ly * LSX;
    const int gy = reflect2(gy0 - PADR + ly, H_DIM);
    const int gx = reflect2(gx0 - PADR + lx, W_DIM);
    const float* gp = src + (size_t)gy * W_DIM + gx;
    float* lp = &tile[i];
#ifdef USE_ASYNC_LDS
    __builtin_amdgcn_global_load_async_to_lds_b32(
        (AS_GLOBAL int*)gp, (AS_SHARED int*)lp, 0, 0);
#else
    *lp = *gp;
#endif
  }
#ifdef USE_ASYNC_LDS
  __builtin_amdgcn_s_wait_asynccnt(0);
#endif
  __syncthreads();

  // ---- Each thread computes one 2x2 RGGB quad ----
  const int qx = tid & 15;      // 16 quads in x
  const int qy = tid >> 4;      // 16 quads in y
  const int c0 = 2 * qx;        // tile col of window col 0 (window center at +2)
  const int r0 = 2 * qy;        // tile row of window row 0

  float w[6][6];
#pragma unroll
  for (int r = 0; r < 6; ++r) {
#pragma unroll
    for (int c = 0; c < 6; c += 2) {
      const float2 v = *(const float2*)&tile[(r0 + r) * LSX + c0 + c];
      w[r][c]     = v.x;
      w[r][c + 1] = v.y;
    }
  }

  // Quad top-left is always an R site (tile origins are even, RGGB):
  // (0,0)=R  (0,1)=Gr  (1,0)=Gb  (1,1)=B
  const float r00 = w[2][2];              // R site
  const float g00 = conv_g (w, 2, 2);
  const float b00 = conv_br(w, 2, 2);

  const float r01 = conv_col(w, 2, 3);    // Gr site
  const float g01 = w[2][3];
  const float b01 = conv_row(w, 2, 3);

  const float r10 = conv_row(w, 3, 2);    // Gb site
  const float g10 = w[3][2];
  const float b10 = conv_col(w, 3, 2);

  const float r11 = conv_br(w, 3, 3);     // B site
  const float g11 = conv_g (w, 3, 3);
  const float b11 = w[3][3];

  // ---- Coalesced float2 stores: [B,3,H,W] ----
  const int gx = gx0 + c0;
  const int gy = gy0 + r0;
  const size_t base = (size_t)img * 3 * plane;
  const size_t off  = base + (size_t)gy * W_DIM + gx;

  *(float2*)&out[off]                       = make_float2(clamp01(r00), clamp01(r01));
  *(float2*)&out[off + W_DIM]               = make_float2(clamp01(r10), clamp01(r11));
  *(float2*)&out[off + plane]               = make_float2(clamp01(g00), clamp01(g01));
  *(float2*)&out[off + plane + W_DIM]       = make_float2(clamp01(g10), clamp01(g11));
  *(float2*)&out[off + 2 * plane]           = make_float2(clamp01(b00), clamp01(b01));
  *(float2*)&out[off + 2 * plane + W_DIM]   = make_float2(clamp01(b10), clamp01(b11));
}

extern "C" void kernel_launch(void* const* d_in, const int* in_sizes, int n_in,
                              void* d_out, int out_size, void* d_ws, size_t ws_size,
                              hipStream_t stream) {
  // d_in[0] = bayer [B,1,2048,2048] fp32; d_in[1..4] = the fixed 5x5 Malvar
  // kernels (compile-time constants in the reference -> folded into the
  // specialized stencil code above, not read at runtime).
  const float* bayer = (const float*)d_in[0];
  float* out = (float*)d_out;
  const int B = in_sizes[0] / (H_DIM * W_DIM);

  dim3 grid(W_DIM / TILE, H_DIM / TILE, B);
  dim3 block(NTHREADS);
  demosaic_malvar_kernel<<<grid, block, 0, stream>>>(bayer, out);
}